// KMadness_87299505258626
// MI455X (gfx1250) — compile-verified
//
#include <hip/hip_runtime.h>
#include <hip/hip_bf16.h>

typedef __attribute__((ext_vector_type(2))) float v2f;
typedef __attribute__((ext_vector_type(8))) float v8f;

#define NROWS 16384
#define KCOLS 128
#define DDIM  256
#define LDSS  129   // padded LDS row stride (129 % 64 == 1 -> conflict-free row scans)

// ---- pre-pass: nsq[k] = ||coords[k]||^2, one wave32 per row (coalesced) ----
__global__ __launch_bounds__(256) void KM_sqnorm_kernel(
    const float* __restrict__ coords, float* __restrict__ nsq) {
    const int lane = threadIdx.x & 31;
    const int row  = blockIdx.x * 8 + (threadIdx.x >> 5);   // 16 blocks * 8 waves = 128 rows
    const float* p = coords + (size_t)row * DDIM;
    float s = 0.f;
    #pragma unroll
    for (int i = 0; i < DDIM / 32; ++i) {                   // 8 coalesced 128B steps
        float v = p[lane + 32 * i];
        s += v * v;
    }
    #pragma unroll
    for (int m = 16; m >= 1; m >>= 1) s += __shfl_xor(s, m, 32);
    if (lane == 0) nsq[row] = s;
}

// ---- main: S = x * coords^T + nsq ; out[n,j] = secondmin_i S[n,i] > S[n,j] --
__global__ __launch_bounds__(256) void KM_secondmin_kernel(
    const float* __restrict__ x,      // (16384, 256)
    const float* __restrict__ coords, // (128, 256)
    const float* __restrict__ nsq,    // (128,)
    float* __restrict__ out)          // (16384, 128) as 0.0/1.0
{
    __shared__ float s_tile[16 * LDSS]; // S tile: 16 rows x 128 cols (padded)
    __shared__ float s_m2[16];          // per-row second-min

    const int tid     = threadIdx.x;
    const int lane    = tid & 31;       // wave32
    const int wave    = tid >> 5;       // 0..7 -> column tile
    const int rowbase = blockIdx.x * 16;
    const int colbase = wave * 16;

    // A 16x4 f32 layout: lane L -> M = L&15, holds K = kb, kb+1 (kb = (L>>4)*2)
    // B  4x16 f32 layout mirrors A: lane L -> N = L&15, holds K = kb, kb+1
    const int mn = lane & 15;
    const int kb = (lane >> 4) << 1;

    const float* arow = x      + (size_t)(rowbase + mn) * DDIM + kb;
    const float* brow = coords + (size_t)(colbase + mn) * DDIM + kb;

    v8f acc = {};
    #pragma unroll 8
    for (int t = 0; t < DDIM / 4; ++t) {        // 64 k-steps of 4
        v2f a = *(const v2f*)(arow + 4 * t);    // 8B-aligned (kb even)
        v2f b = *(const v2f*)(brow + 4 * t);
        // (neg_a, A, neg_b, B, c_mod, C, reuse_a, reuse_b)
        acc = __builtin_amdgcn_wmma_f32_16x16x4_f32(
            false, a, false, b, (short)0, acc, false, false);
    }

    // C/D layout: lane L -> col N = L&15 (constant per lane); VGPR r -> row r + 8*(L>>4)
    const float bias = nsq[colbase + mn];       // + ||c_col||^2
    const int   rb   = (lane >> 4) << 3;
    const int   col  = colbase + mn;
    #pragma unroll
    for (int r = 0; r < 8; ++r)
        s_tile[(rb + r) * LDSS + col] = acc[r] + bias;

    __syncthreads();

    // per-row (min1, min2): wave 0 only; lane = (row, half), branchless scan of
    // 64 cols each, then one xor-16 shuffle merge of the two halves.
    if (wave == 0) {
        const int   rrow  = lane & 15;
        const int   cbase = (lane >> 4) << 6;   // 0 or 64
        const float* rowp = s_tile + rrow * LDSS + cbase;
        float m1 = __builtin_inff(), m2 = __builtin_inff();
        #pragma unroll 8
        for (int c = 0; c < 64; ++c) {
            float v = rowp[c];
            m2 = fminf(m2, fmaxf(m1, v));       // branchless; duplicates behave
            m1 = fminf(m1, v);                  // like sorted second element
        }
        float o1 = __shfl_xor(m1, 16, 32);
        float o2 = __shfl_xor(m2, 16, 32);
        float mm2 = fminf(fmaxf(m1, o1), fminf(m2, o2));
        if (lane < 16) s_m2[rrow] = mm2;
    }
    __syncthreads();

    // out[n, j] = (secondmin_i S[n,i]) > S[n,j]; coalesced 128-float rows
    #pragma unroll
    for (int e = 0; e < 8; ++e) {
        int idx = tid + 256 * e;                // 0..2047 over the 16x128 tile
        int r = idx >> 7;
        int c = idx & 127;
        out[(size_t)(rowbase + r) * KCOLS + c] =
            (s_m2[r] > s_tile[r * LDSS + c]) ? 1.0f : 0.0f;
    }
}

extern "C" void kernel_launch(void* const* d_in, const int* in_sizes, int n_in,
                              void* d_out, int out_size, void* d_ws, size_t ws_size,
                              hipStream_t stream) {
    const float* x      = (const float*)d_in[0];   // (16384, 256) f32
    const float* coords = (const float*)d_in[1];   // (128, 256)   f32
    float*       out    = (float*)d_out;           // (16384, 128) bool -> f32 0/1
    float*       nsq    = (float*)d_ws;            // 128 floats scratch

    KM_sqnorm_kernel<<<16, 256, 0, stream>>>(coords, nsq);
    KM_secondmin_kernel<<<NROWS / 16, 256, 0, stream>>>(x, coords, nsq, out);
}